// MHSA_58918361366874
// MI455X (gfx1250) — compile-verified
//
#include <hip/hip_runtime.h>

typedef float v2f __attribute__((ext_vector_type(2)));
typedef float v8f __attribute__((ext_vector_type(8)));
typedef float v4fv __attribute__((ext_vector_type(4)));

#define NH 4
#define HD 128
#define HW 1024
#define IND 512
#define NB 16
#define SCALE 0.08838834764831845f  // 128^-0.5

#define XPITCH 72  // LDS pitch for 64-wide x slice (bank-conflict-free)
#define SPITCH 72  // LDS pitch for 64-wide sim strip

static __device__ __forceinline__ v8f wmma_f32(v2f a, v2f b, v8f c) {
  // D = A(16x4 f32) * B(4x16 f32) + C(16x16 f32)
  return __builtin_amdgcn_wmma_f32_16x16x4_f32(false, a, false, b, (short)0, c,
                                               false, false);
}

static __device__ __forceinline__ v8f v8f_zero() {
  v8f z;
#pragma unroll
  for (int i = 0; i < 8; i++) z[i] = 0.0f;
  return z;
}

// -------------------------------------------------------------------------
// Kernel 1: QKV projection GEMM with fused q-scale and pos-emb fold into k'.
// Per batch: C[o,p] = sum_c W[o,c] * x[b,c,p]
// Block tile: 256 o x 64 p. x slice (512x64, 144KB) staged in LDS and
// reused by all 16 o-tiles; W rows touched exactly once per block.
// Grid: b(16) x og(6) x pg(16) = 1536 blocks, 256 threads (8 waves).
// Wave: 2 o-tiles x 4 p-tiles (8 accumulators).
//   q  -> qT[bh][d][i]  (val * SCALE)
//   k' -> kT[bh][d][j]  (val + ph[j/32,d] + pw[j%32,d])
//   v  -> v [bh][j][d]
// -------------------------------------------------------------------------
__global__ __launch_bounds__(256) void qkv_kernel(
    const float* __restrict__ x, const float* __restrict__ w,
    const float* __restrict__ ph, const float* __restrict__ pw,
    float* __restrict__ qT, float* __restrict__ kT, float* __restrict__ vout) {
  __shared__ float xlds[IND * XPITCH];  // 512 x 64 (pitch 72) = 144 KB

  const int bid = blockIdx.x;
  const int b = bid / 96;
  const int rem = bid % 96;
  const int og = rem / 16;   // 0..5 : 256-wide o group
  const int pg = rem % 16;   // 0..15: 64-wide p group
  const int p0 = pg * 64;
  const int tid = threadIdx.x;
  const int wv = tid >> 5;
  const int lane = tid & 31;
  const int m = lane & 15;
  const int kh = (lane >> 4) << 1;  // 0 or 2

  // Cooperative stage of x[b][0..512][p0..p0+64) with float4 loads.
  {
    const float* xb = x + (size_t)b * IND * HW + p0;
    for (int idx = tid; idx < IND * 16; idx += 256) {
      const int row = idx >> 4;
      const int c4 = (idx & 15) << 2;
      const v4fv val = *(const v4fv*)(xb + (size_t)row * HW + c4);
      xlds[row * XPITCH + c4 + 0] = val[0];
      xlds[row * XPITCH + c4 + 1] = val[1];
      xlds[row * XPITCH + c4 + 2] = val[2];
      xlds[row * XPITCH + c4 + 3] = val[3];
    }
  }
  __syncthreads();

  v8f acc[2][4];
#pragma unroll
  for (int o2 = 0; o2 < 2; o2++)
#pragma unroll
    for (int t = 0; t < 4; t++) acc[o2][t] = v8f_zero();

  const int otg0 = og * 16 + wv * 2;  // first of 2 global o-tiles for wave

  for (int k0 = 0; k0 < IND; k0 += 4) {
    const int kc = k0 + kh;
    v2f a[2];
#pragma unroll
    for (int o2 = 0; o2 < 2; o2++) {
      const size_t wrow = (size_t)((otg0 + o2) * 16 + m) * IND;
      a[o2].x = w[wrow + kc];
      a[o2].y = w[wrow + kc + 1];
    }
#pragma unroll
    for (int t = 0; t < 4; t++) {
      v2f bb;
      bb.x = xlds[kc * XPITCH + t * 16 + m];
      bb.y = xlds[(kc + 1) * XPITCH + t * 16 + m];
#pragma unroll
      for (int o2 = 0; o2 < 2; o2++)
        acc[o2][t] = wmma_f32(a[o2], bb, acc[o2][t]);
    }
  }

#pragma unroll
  for (int o2 = 0; o2 < 2; o2++) {
    const int ob = (otg0 + o2) * 16;
    const int sect = ob >> 9;       // 0=q 1=k 2=v (uniform per tile)
    const int h = (ob & 511) >> 7;  // head
    const int dbase = ob & 127;
    const int bh = b * NH + h;
#pragma unroll
    for (int t = 0; t < 4; t++) {
      const int p = p0 + t * 16 + m;
#pragma unroll
      for (int r = 0; r < 8; r++) {
        const int d = dbase + r + ((lane >> 4) << 3);
        const float val = acc[o2][t][r];
        if (sect == 0) {
          qT[((size_t)bh * HD + d) * HW + p] = val * SCALE;
        } else if (sect == 1) {
          const float e = ph[(p >> 5) * HD + d] + pw[(p & 31) * HD + d];
          kT[((size_t)bh * HD + d) * HW + p] = val + e;
        } else {
          vout[((size_t)bh * HW + p) * HD + d] = val;
        }
      }
    }
  }
}

// -------------------------------------------------------------------------
// Kernel 2: attention for one (b, h, 64-row i-block).
// Phase 1: sim(64 x 1024) = Q_blk * K'^T via WMMA -> LDS simT[j][i] (288KB).
//          K' is streamed exactly once per block.
// Phase 2: row softmax, 4-lane shuffle reductions.
// Phase 3: out^T(d x i) = V^T * attn^T via WMMA; V streamed once per block;
//          each V load amortized over 4 i-tiles. Coalesced final stores.
// Grid: b(16) x h(4) x ib(16) = 1024 blocks, 256 threads (8 waves).
// -------------------------------------------------------------------------
__global__ __launch_bounds__(256) void attn_kernel(
    const float* __restrict__ qT, const float* __restrict__ kT,
    const float* __restrict__ v, float* __restrict__ out) {
  __shared__ float simT[HW * SPITCH];  // [j][i], 1024 x 64 (pitch 72) = 288 KB

  const int bid = blockIdx.x;
  const int b = bid >> 6;
  const int h = (bid >> 4) & 3;
  const int ib = bid & 15;
  const int bh = b * NH + h;
  const int i0 = ib * 64;
  const int tid = threadIdx.x;
  const int wv = tid >> 5;
  const int lane = tid & 31;
  const int m = lane & 15;
  const int kh = (lane >> 4) << 1;

  const float* qb = qT + (size_t)bh * HD * HW;
  const float* kb = kT + (size_t)bh * HD * HW;
  const float* vb = v + (size_t)bh * HW * HD;

  // ---- Phase 1: sim strip; wave covers 8 j-tiles as 4 pairs x 4 i-tiles ----
  for (int g = 0; g < 4; g++) {
    const int jt0 = wv * 8 + g * 2;  // pair of j-tiles
    v8f acc[4][2];
#pragma unroll
    for (int it = 0; it < 4; it++)
#pragma unroll
      for (int jj = 0; jj < 2; jj++) acc[it][jj] = v8f_zero();

    for (int k0 = 0; k0 < HD; k0 += 4) {
      const int kc = k0 + kh;
      v2f b0, b1;  // B = K'^T (K=d, N=j) == kT[d][j] row-major
      b0.x = kb[(size_t)kc * HW + jt0 * 16 + m];
      b0.y = kb[(size_t)(kc + 1) * HW + jt0 * 16 + m];
      b1.x = kb[(size_t)kc * HW + (jt0 + 1) * 16 + m];
      b1.y = kb[(size_t)(kc + 1) * HW + (jt0 + 1) * 16 + m];
#pragma unroll
      for (int it = 0; it < 4; it++) {
        v2f a;  // A = Q (M=i, K=d) from qT[d][i]
        a.x = qb[(size_t)kc * HW + i0 + it * 16 + m];
        a.y = qb[(size_t)(kc + 1) * HW + i0 + it * 16 + m];
        acc[it][0] = wmma_f32(a, b0, acc[it][0]);
        acc[it][1] = wmma_f32(a, b1, acc[it][1]);
      }
    }

#pragma unroll
    for (int jj = 0; jj < 2; jj++) {
      const int j16 = (jt0 + jj) * 16;
#pragma unroll
      for (int it = 0; it < 4; it++) {
#pragma unroll
        for (int r = 0; r < 8; r++) {
          const int il = it * 16 + r + ((lane >> 4) << 3);
          simT[(j16 + m) * SPITCH + il] = acc[it][jj][r];
        }
      }
    }
  }
  __syncthreads();

  // ---- Phase 2: softmax along j for each of 64 rows (i) ----
  {
    const int row = tid >> 2;  // 0..63, handled by a 4-lane group
    const int c0 = tid & 3;
    float mx = -3.4e38f;
    for (int c = c0; c < HW; c += 4) mx = fmaxf(mx, simT[c * SPITCH + row]);
#pragma unroll
    for (int off = 2; off; off >>= 1) mx = fmaxf(mx, __shfl_xor(mx, off, 4));
    float sum = 0.0f;
    for (int c = c0; c < HW; c += 4) {
      const float e = __expf(simT[c * SPITCH + row] - mx);
      simT[c * SPITCH + row] = e;
      sum += e;
    }
#pragma unroll
    for (int off = 2; off; off >>= 1) sum += __shfl_xor(sum, off, 4);
    const float inv = 1.0f / sum;
    for (int c = c0; c < HW; c += 4) simT[c * SPITCH + row] *= inv;
  }
  __syncthreads();

  // ---- Phase 3: out^T (M=d tile per wave, N=i x4 tiles, K=j over 1024) ----
  const int dt = wv;
  v8f oacc[4];
#pragma unroll
  for (int it = 0; it < 4; it++) oacc[it] = v8f_zero();

  for (int j0 = 0; j0 < HW; j0 += 4) {
    const int kc = j0 + kh;
    v2f a;  // A = V^T (M=d, K=j) from v[j][d]
    a.x = vb[(size_t)kc * HD + dt * 16 + m];
    a.y = vb[(size_t)(kc + 1) * HD + dt * 16 + m];
#pragma unroll
    for (int it = 0; it < 4; it++) {
      v2f bb;  // B = attn^T (K=j, N=i) from simT[j][i]
      bb.x = simT[kc * SPITCH + it * 16 + m];
      bb.y = simT[(kc + 1) * SPITCH + it * 16 + m];
      oacc[it] = wmma_f32(a, bb, oacc[it]);
    }
  }

  // out layout: [B][nh][hd][hw] flat -> lanes vary i: coalesced
#pragma unroll
  for (int it = 0; it < 4; it++) {
#pragma unroll
    for (int r = 0; r < 8; r++) {
      const int d = dt * 16 + r + ((lane >> 4) << 3);
      out[((size_t)bh * HD + d) * HW + i0 + it * 16 + m] = oacc[it][r];
    }
  }
}

extern "C" void kernel_launch(void* const* d_in, const int* in_sizes, int n_in,
                              void* d_out, int out_size, void* d_ws,
                              size_t ws_size, hipStream_t stream) {
  const float* x = (const float*)d_in[0];   // [16,512,32,32]
  const float* w = (const float*)d_in[1];   // [1536,512]
  const float* ph = (const float*)d_in[2];  // [32,128]
  const float* pw = (const float*)d_in[3];  // [32,128]
  float* out = (float*)d_out;               // [16,512,256,4] flat

  const size_t seg = (size_t)NB * NH * HW * HD;  // 8,388,608 floats (32 MB)
  float* qT = (float*)d_ws;
  float* kT = qT + seg;
  float* v = kT + seg;

  qkv_kernel<<<NB * 6 * 16, 256, 0, stream>>>(x, w, ph, pw, qT, kT, v);
  attn_kernel<<<NB * NH * 16, 256, 0, stream>>>(qT, kT, v, out);
}